// ExplicitVolume_59236188947222
// MI455X (gfx1250) — compile-verified
//
#include <hip/hip_runtime.h>
#include <math.h>

#define SLEN  128
#define NBATCH 16

typedef float v2f __attribute__((ext_vector_type(2)));
typedef float v8f __attribute__((ext_vector_type(8)));

__device__ __forceinline__ float lin_coord(int t) {
    // linspace(-1, 1, 128)
    return fmaf((float)t, 2.0f / 127.0f, -1.0f);
}

__device__ __forceinline__ int clamp127(int v) {
    v = v < 0 ? 0 : v;
    return v > 127 ? 127 : v;
}

__global__ __launch_bounds__(128)
void vol_rot_proj_kernel(const float* __restrict__ rotmat,
                         const float* __restrict__ scalar,
                         const float* __restrict__ vol,
                         float* __restrict__ out)
{
    const int b = blockIdx.x >> 7;      // batch
    const int i = blockIdx.x & 127;     // output row (grid y channel)
    const int j = threadIdx.x;          // output col (grid x channel)
    const int lane = threadIdx.x & 31;

    const float* R = rotmat + b * 9;    // R[c][d] = R[c*3 + d]

    // ---- Base rotation (x,y part) via V_WMMA_F32_16X16X4_F32 ----
    // D[d][p] = sum_c A[d][c] * B[c][p],  A = R^T (rows d=0..2), B = pixel coords.
    // A 16x4 layout: v0 = K0 (lanes 0-15) | K2 (lanes 16-31); v1 = K1 | K3.
    v2f A;
    A[0] = (lane < 3) ? R[0 * 3 + lane] : 0.0f;   // A[d][0] = R[0][d]
    A[1] = (lane < 3) ? R[1 * 3 + lane] : 0.0f;   // A[d][1] = R[1][d]

    const float yg = lin_coord(i);
    const int jw = j & ~31;             // wave's 32-pixel base
    const bool lo = lane < 16;
    const int sel = lane & 15;

    // B 4x16 layout (mirrors A): v0 = K0 row | K2 row; v1 = K1 row | K3 row.
    v2f B1, B2;
    B1[0] = lo ? lin_coord(jw + sel)      : 0.0f;   // x of pixels 0..15
    B1[1] = lo ? yg                       : 0.0f;   // y (uniform)
    B2[0] = lo ? lin_coord(jw + 16 + sel) : 0.0f;   // x of pixels 16..31
    B2[1] = lo ? yg                       : 0.0f;

    v8f Cz = {};
    v8f D1 = __builtin_amdgcn_wmma_f32_16x16x4_f32(false, A, false, B1,
                                                   (short)0, Cz, false, false);
    v8f D2 = __builtin_amdgcn_wmma_f32_16x16x4_f32(false, A, false, B2,
                                                   (short)0, Cz, false, false);

    // D VGPR d, lanes 0-15 hold component d of pixel = lane.
    // Route pixel p's (gx,gy,gz) base to lane p (lanes 16-31 come from D2).
    float s1x = __shfl(D1[0], sel, 32), s2x = __shfl(D2[0], sel, 32);
    float s1y = __shfl(D1[1], sel, 32), s2y = __shfl(D2[1], sel, 32);
    float s1z = __shfl(D1[2], sel, 32), s2z = __shfl(D2[2], sel, 32);
    const float bx = lo ? s1x : s2x;
    const float by = lo ? s1y : s2y;
    const float bz = lo ? s1z : s2z;

    // z-direction step coefficients: row 2 of R
    const float dzx = R[6], dzy = R[7], dzz = R[8];

    float acc = 0.0f;
#pragma unroll 4
    for (int k = 0; k < SLEN; ++k) {
        const float zg = lin_coord(k);
        const float gx = fmaf(zg, dzx, bx);
        const float gy = fmaf(zg, dzy, by);
        const float gz = fmaf(zg, dzz, bz);

        // align_corners=True unnormalize: u = (g+1)*63.5
        const float ux = fmaf(gx, 63.5f, 63.5f);
        const float uy = fmaf(gy, 63.5f, 63.5f);
        const float uz = fmaf(gz, 63.5f, 63.5f);

        const float fx = floorf(ux); const int ix = (int)fx; const float tx = ux - fx;
        const float fy = floorf(uy); const int iy = (int)fy; const float ty = uy - fy;
        const float fz = floorf(uz); const int iz = (int)fz; const float tz = uz - fz;

        // zero-padding: invalid corner => weight 0; index clamped for safe load
        const float wx0 = ((unsigned)ix       < 128u) ? (1.0f - tx) : 0.0f;
        const float wx1 = ((unsigned)(ix + 1) < 128u) ? tx          : 0.0f;
        const float wy0 = ((unsigned)iy       < 128u) ? (1.0f - ty) : 0.0f;
        const float wy1 = ((unsigned)(iy + 1) < 128u) ? ty          : 0.0f;
        const float wz0 = ((unsigned)iz       < 128u) ? (1.0f - tz) : 0.0f;
        const float wz1 = ((unsigned)(iz + 1) < 128u) ? tz          : 0.0f;

        const int ix0 = clamp127(ix), ix1 = clamp127(ix + 1);
        const int iy0 = clamp127(iy), iy1 = clamp127(iy + 1);
        const int iz0 = clamp127(iz), iz1 = clamp127(iz + 1);

        const float* p00 = vol + ((iz0 << 7) + iy0 << 7);
        const float* p01 = vol + ((iz0 << 7) + iy1 << 7);
        const float* p10 = vol + ((iz1 << 7) + iy0 << 7);
        const float* p11 = vol + ((iz1 << 7) + iy1 << 7);
        const float v000 = p00[ix0], v001 = p00[ix1];
        const float v010 = p01[ix0], v011 = p01[ix1];
        const float v100 = p10[ix0], v101 = p10[ix1];
        const float v110 = p11[ix0], v111 = p11[ix1];

        const float sx00 = fmaf(wx1, v001, wx0 * v000);
        const float sx01 = fmaf(wx1, v011, wx0 * v010);
        const float sx10 = fmaf(wx1, v101, wx0 * v100);
        const float sx11 = fmaf(wx1, v111, wx0 * v110);
        const float sy0  = fmaf(wy1, sx01, wy0 * sx00);
        const float sy1  = fmaf(wy1, sx11, wy0 * sx10);
        acc = fmaf(wz0, sy0, acc);
        acc = fmaf(wz1, sy1, acc);
    }

    const float scale = expf(scalar[1]);
    const int idx = (b * SLEN + i) * SLEN + j;
    const float r = scale * acc;
    out[idx] = r;                                  // projection
    out[NBATCH * SLEN * SLEN + idx] = r;           // projection_clean (identical)
}

extern "C" void kernel_launch(void* const* d_in, const int* in_sizes, int n_in,
                              void* d_out, int out_size, void* d_ws, size_t ws_size,
                              hipStream_t stream) {
    (void)in_sizes; (void)n_in; (void)out_size; (void)d_ws; (void)ws_size;
    const float* rotmat = (const float*)d_in[0];   // (16,3,3) f32
    const float* scalar = (const float*)d_in[1];   // (2,) f32
    const float* vol    = (const float*)d_in[2];   // (128,128,128) f32
    float* out = (float*)d_out;                    // 2 * 16*128*128 f32

    dim3 grid(NBATCH * SLEN);   // one block per (b, i)
    dim3 block(SLEN);           // thread = j, 4 wave32s
    vol_rot_proj_kernel<<<grid, block, 0, stream>>>(rotmat, scalar, vol, out);
}